// test1_encoder_50895362458301
// MI455X (gfx1250) — compile-verified
//
#include <hip/hip_runtime.h>

typedef float v2f __attribute__((ext_vector_type(2)));
typedef float v8f __attribute__((ext_vector_type(8)));

#define B_N  1024
#define T_N  512
#define HH   8
#define G3   24   // 3*H valid gate columns
#define G3S  32   // padded row stride for xg staging (128B-aligned rows)
#define TPW  8    // timesteps processed per wave in the projection kernel

// ---------------------------------------------------------------------------
// Kernel 1: input projection  xg[t][b][0..23] = x[b][t][:] @ Wih^T
// via V_WMMA_F32_16X16X4_F32.  One wave computes 16(batch-rows) x 32 tiles
// for TPW consecutive timesteps, reusing the Wih B-fragments.
// RS/OFF/IN are compile-time so the K-padding predicates fold away:
//   bbox (IN=4): unconditional b64 A-loads, zero branches in the loop
//   flow (IN=2): single predicated b64 A-load (lanes 16-31 statically zero)
// N padded 24 -> 32 so all stores are unconditional.
// x element (b,t,i) lives at x[(b*T + t)*RS + OFF + i].
// ---------------------------------------------------------------------------
template <int RS, int OFF, int IN>
__global__ __launch_bounds__(256) void xproj_wmma_kernel(
    const float* __restrict__ x, const float* __restrict__ Wih,
    float* __restrict__ xg)
{
  const int lane = threadIdx.x & 31;
  const int wid  = blockIdx.x * (blockDim.x >> 5) + (threadIdx.x >> 5);

  const int tilesPerT = B_N / 16;            // 64
  const int t0 = (wid / tilesPerT) * TPW;
  const int b0 = (wid % tilesPerT) * 16;

  const int m  = lane & 15;                  // matrix row within tile
  const int kh = lane >> 4;                  // K half: lanes16-31 hold K=2,3
  const int k0 = kh * 2;

  // B fragments: Wih^T (4x16), two N tiles covering N=0..15 / N=16..31.
  // B[k][n] = Wih[n*IN + k]; zero-pad K>=IN and N>=24.  Loop-invariant.
  v2f bf0 = {}, bf1 = {};
  const int n  = m;
  const int n1 = m + 16;
  if (k0 < IN) {
    bf0.x = Wih[n * IN + k0];
    bf0.y = (k0 + 1 < IN) ? Wih[n * IN + k0 + 1] : 0.0f;
    if (n1 < G3) {
      bf1.x = Wih[n1 * IN + k0];
      bf1.y = (k0 + 1 < IN) ? Wih[n1 * IN + k0 + 1] : 0.0f;
    }
  }

  // Per-lane source pointer for this batch row; advance by RS per timestep.
  // (row records are 8B-aligned for both layouts: RS*4 and OFF*4 are
  //  multiples of 8, so v2f (b64) loads are legal.)
  const float* src = x + ((long)(b0 + m) * T_N + t0) * RS + OFF + k0;

  for (int i = 0; i < TPW; ++i) {
    const int t = t0 + i;

    // A fragment: 16x4 f32 (2 VGPRs/lane per ISA layout)
    v2f a = {};
    if (k0 + 1 < IN) {                 // statically true/false per half
      a = *(const v2f*)(src + (long)i * RS);
    } else if (k0 < IN) {
      a.x = src[(long)i * RS];
    }

    v8f zero = {};
    // (neg_a, A, neg_b, B, c_mod, C, reuse_a, reuse_b)
    v8f c0 = __builtin_amdgcn_wmma_f32_16x16x4_f32(false, a, false, bf0,
                                                   (short)0, zero, false, false);
    v8f c1 = __builtin_amdgcn_wmma_f32_16x16x4_f32(false, a, false, bf1,
                                                   (short)0, zero, false, false);

    // C layout: VGPR v, lanes0-15 -> (row v, col n); lanes16-31 -> (row v+8).
    // Padded stride => every lane stores 16 dwords unconditionally.
    float* outp = xg + ((long)t * B_N + b0) * G3S + n;
#pragma unroll
    for (int v = 0; v < 8; ++v) {
      const int mr = v + 8 * kh;
      outp[(long)mr * G3S]      = c0[v];
      outp[(long)mr * G3S + 16] = c1[v];
    }
  }
}

// ---------------------------------------------------------------------------
// Kernel 2: GRU scan over T.  8 lanes per batch row (lane j = hidden unit j),
// h kept in registers, h[j'] gathered via lane permute within the 8-lane group.
// xg layout [T][B][32] (L2-resident, 64 MB < 192 MB L2) gives coalesced,
// 128B-aligned row reads per step.
// combine==0: out = h_final          (bbox pass, writes scratch)
// combine==1: out = 0.5*(h + other)  (flow pass, writes final feature)
// ---------------------------------------------------------------------------
__global__ __launch_bounds__(256) void gru_scan_kernel(
    const float* __restrict__ xg,  const float* __restrict__ Whh,
    const float* __restrict__ bih, const float* __restrict__ bhh,
    const float* __restrict__ h_other, float* __restrict__ out, int combine)
{
  const int tid = blockIdx.x * blockDim.x + threadIdx.x;
  const int j = tid & 7;        // hidden unit
  const int b = tid >> 3;       // batch row

  float wr[HH], wz[HH], wn[HH];
#pragma unroll
  for (int k = 0; k < HH; ++k) {
    wr[k] = Whh[(j     ) * HH + k];
    wz[k] = Whh[(j +  8) * HH + k];
    wn[k] = Whh[(j + 16) * HH + k];
  }
  const float br   = bih[j]      + bhh[j];
  const float bz   = bih[j +  8] + bhh[j +  8];
  const float bn_i = bih[j + 16];
  const float bn_h = bhh[j + 16];

  float h = 0.0f;
  for (int t = 0; t < T_N; ++t) {
    const float* row = xg + ((long)t * B_N + b) * G3S;
    const float xr = row[j], xz = row[j + 8], xn = row[j + 16];

    float gr = 0.0f, gz = 0.0f, gn = 0.0f;
#pragma unroll
    for (int k = 0; k < HH; ++k) {
      const float hk = __shfl(h, k, 8);   // h[k] from this row's lane group
      gr = fmaf(hk, wr[k], gr);
      gz = fmaf(hk, wz[k], gz);
      gn = fmaf(hk, wn[k], gn);
    }
    const float r = 1.0f / (1.0f + __expf(-(xr + gr + br)));
    const float z = 1.0f / (1.0f + __expf(-(xz + gz + bz)));
    const float n = tanhf(xn + bn_i + r * (gn + bn_h));
    h = (1.0f - z) * n + z * h;
  }

  if (combine)
    out[(long)b * HH + j] = 0.5f * (h + h_other[(long)b * HH + j]);
  else
    out[(long)b * HH + j] = h;
}

// ---------------------------------------------------------------------------
// inputs: 0 bbox [B,T,4], 1 flow [B,T,5,5,2], 2 Wih_b, 3 Whh_b, 4 bih_b,
//         5 bhh_b, 6 Wih_f, 7 Whh_f, 8 bih_f, 9 bhh_f
// ws: xg staging (T*B*32 f32 = 64 MB, reused for both GRUs) + h_bbox (32 KB)
// ---------------------------------------------------------------------------
extern "C" void kernel_launch(void* const* d_in, const int* in_sizes, int n_in,
                              void* d_out, int out_size, void* d_ws, size_t ws_size,
                              hipStream_t stream) {
  const float* bbox  = (const float*)d_in[0];
  const float* flow  = (const float*)d_in[1];
  const float* Wih_b = (const float*)d_in[2];
  const float* Whh_b = (const float*)d_in[3];
  const float* bih_b = (const float*)d_in[4];
  const float* bhh_b = (const float*)d_in[5];
  const float* Wih_f = (const float*)d_in[6];
  const float* Whh_f = (const float*)d_in[7];
  const float* bih_f = (const float*)d_in[8];
  const float* bhh_f = (const float*)d_in[9];
  float* out = (float*)d_out;

  float* xg = (float*)d_ws;
  float* hb = xg + (size_t)T_N * B_N * G3S;

  // tiles = (T/TPW) * (B/16) waves, 8 waves per 256-thread block
  const int k1_blocks = ((T_N / TPW) * (B_N / 16)) / 8;   // 512
  const int k2_blocks = (B_N * HH) / 256;                 // 32

  // bbox GRU: In=4 (row stride 4, offset 0)
  xproj_wmma_kernel<4, 0, 4><<<k1_blocks, 256, 0, stream>>>(bbox, Wih_b, xg);
  gru_scan_kernel<<<k2_blocks, 256, 0, stream>>>(xg, Whh_b, bih_b, bhh_b,
                                                 nullptr, hb, 0);
  // flow GRU: center pixel -> In=2 (row stride 50, offset (2*5+2)*2 = 24)
  xproj_wmma_kernel<50, 24, 2><<<k1_blocks, 256, 0, stream>>>(flow, Wih_f, xg);
  gru_scan_kernel<<<k2_blocks, 256, 0, stream>>>(xg, Whh_f, bih_f, bhh_f,
                                                 hb, out, 1);
}